// DrugEncoder_42606075576811
// MI455X (gfx1250) — compile-verified
//
#include <hip/hip_runtime.h>
#include <hip/hip_bf16.h>

typedef __attribute__((ext_vector_type(2))) float v2f;
typedef __attribute__((ext_vector_type(8))) float v8f;

#define HID 128
#define EDGE_DIM 16
#define BN_EPS 1e-5f

// ---------------------------------------------------------------------------
// zero fill
// ---------------------------------------------------------------------------
__global__ void zero_f32_kernel(float* __restrict__ p, long long n) {
  long long i = (long long)blockIdx.x * blockDim.x + threadIdx.x;
  if (i < n) p[i] = 0.0f;
}

// ---------------------------------------------------------------------------
// Edge phase: agg[dst] += relu(h[src] + edge_attr @ We + be)
// One wave (32 lanes) per edge, C/32 columns per lane. We (16 x C) staged in
// LDS; edge_attr row broadcast via wave32 shuffles. Scatter uses HW f32
// atomics (global_atomic_add_f32).
// ---------------------------------------------------------------------------
__global__ __launch_bounds__(256)
void gine_edge_kernel(const float* __restrict__ h,     // [N, C]
                      const float* __restrict__ ea,    // [E, 16]
                      const int*   __restrict__ srcI,  // edge_index[0]
                      const int*   __restrict__ dstI,  // edge_index[1]
                      const float* __restrict__ ew,    // [16, C]
                      const float* __restrict__ eb,    // [C]
                      float* __restrict__ agg,         // [N, C] (pre-zeroed)
                      int C, int nedges)
{
  __shared__ float Ws[EDGE_DIM * HID];
  __shared__ float Bs[HID];
  for (int i = threadIdx.x; i < EDGE_DIM * C; i += 256) Ws[i] = ew[i];
  for (int i = threadIdx.x; i < C; i += 256) Bs[i] = eb[i];
  __syncthreads();

  const int lane = threadIdx.x & 31;
  const int wave = threadIdx.x >> 5;
  const int wstride = gridDim.x * 8;
  const int cpl = C >> 5;  // columns per lane: 2 (C=64) or 4 (C=128)

  for (int e = blockIdx.x * 8 + wave; e < nedges; e += wstride) {
    const int s = srcI[e];
    const int d = dstI[e];
    // lanes 0..15 hold the 16 edge features; broadcast with shuffles
    float eav = (lane < EDGE_DIM) ? ea[(size_t)e * EDGE_DIM + lane] : 0.0f;
    for (int j = 0; j < cpl; ++j) {
      const int col = lane + 32 * j;
      float acc = Bs[col];
#pragma unroll
      for (int k = 0; k < EDGE_DIM; ++k) {
        float a = __shfl(eav, k);
        acc = fmaf(a, Ws[k * C + col], acc);
      }
      float m = h[(size_t)s * C + col] + acc;
      m = m > 0.0f ? m : 0.0f;
      __hip_atomic_fetch_add(&agg[(size_t)d * C + col], m,
                             __ATOMIC_RELAXED, __HIP_MEMORY_SCOPE_AGENT);
    }
  }
}

// ---------------------------------------------------------------------------
// Node MLP phase (WMMA f32 16x16x4):
//   z = hprev + agg
//   t = relu( bn_in(z @ w1 + b1) )           (inner BN folded: t = relu(acc*s1+c1))
//   o = relu( bn_out(t @ w2 + b2) ) [+ resid]
// Block = 256 threads (8 waves) owns a 16-node tile. Wave w owns columns
// [16w, 16w+16). Kin/4 WMMA steps for GEMM 1, 32 for GEMM 2, per wave; t staged through LDS.
// ---------------------------------------------------------------------------
__global__ __launch_bounds__(256)
void gine_mlp_wmma_kernel(const float* __restrict__ hprev,  // [N, Kin]
                          const float* __restrict__ agg,    // [N, Kin]
                          int Kin,
                          const float* __restrict__ w1,     // [Kin, 128]
                          const float* __restrict__ b1,
                          const float* __restrict__ ig, const float* __restrict__ ib,
                          const float* __restrict__ im, const float* __restrict__ iv,
                          const float* __restrict__ w2,     // [128, 128]
                          const float* __restrict__ b2,
                          const float* __restrict__ og, const float* __restrict__ ob,
                          const float* __restrict__ om, const float* __restrict__ ov,
                          const float* __restrict__ resid,  // [N,128] or nullptr
                          float* __restrict__ hout)         // [N, 128]
{
  __shared__ float zs[16 * HID];  // z tile  (16 x Kin, Kin <= 128)
  __shared__ float ts[16 * HID];  // t tile  (16 x 128)

  const int tid  = threadIdx.x;
  const int lane = tid & 31;
  const int wave = tid >> 5;            // 0..7
  const int row0 = blockIdx.x * 16;     // first node of tile

  // stage z = hprev + agg
  for (int i = tid; i < 16 * Kin; i += 256) {
    const int r = i / Kin, c = i % Kin;
    const size_t g = (size_t)(row0 + r) * Kin + c;
    zs[i] = hprev[g] + agg[g];
  }
  __syncthreads();

  const int nloc = lane & 15;           // column within wave tile == A row (M)
  const int half = lane >> 4;           // lane half selects K pair
  const int koff = half * 2;
  const int n    = wave * 16 + nloc;    // global feature column 0..127

  // ---- GEMM 1: z (16 x Kin) @ w1 (Kin x 128) ----
  v8f acc = {};
  for (int kk = 0; kk < Kin; kk += 4) {
    v2f a, b;
    a.x = zs[nloc * Kin + kk + koff + 0];
    a.y = zs[nloc * Kin + kk + koff + 1];
    b.x = w1[(size_t)(kk + koff + 0) * HID + n];
    b.y = w1[(size_t)(kk + koff + 1) * HID + n];
    acc = __builtin_amdgcn_wmma_f32_16x16x4_f32(false, a, false, b,
                                                (short)0, acc, false, false);
  }

  // inner BN + ReLU folded into per-lane (per-feature) scale/shift
  {
    const float s1 = ig[n] * rsqrtf(iv[n] + BN_EPS);
    const float c1 = (b1[n] - im[n]) * s1 + ib[n];
#pragma unroll
    for (int r = 0; r < 8; ++r) {
      float t = acc[r] * s1 + c1;     // D layout: row = r + 8*half, col = n
      t = t > 0.0f ? t : 0.0f;
      ts[(r + half * 8) * HID + n] = t;
    }
  }
  __syncthreads();

  // ---- GEMM 2: t (16 x 128) @ w2 (128 x 128) ----
  v8f acc2 = {};
  for (int kk = 0; kk < HID; kk += 4) {
    v2f a, b;
    a.x = ts[nloc * HID + kk + koff + 0];
    a.y = ts[nloc * HID + kk + koff + 1];
    b.x = w2[(size_t)(kk + koff + 0) * HID + n];
    b.y = w2[(size_t)(kk + koff + 1) * HID + n];
    acc2 = __builtin_amdgcn_wmma_f32_16x16x4_f32(false, a, false, b,
                                                 (short)0, acc2, false, false);
  }

  // outer BN + ReLU (+ residual) and store
  {
    const float s2 = og[n] * rsqrtf(ov[n] + BN_EPS);
    const float c2 = (b2[n] - om[n]) * s2 + ob[n];
#pragma unroll
    for (int r = 0; r < 8; ++r) {
      const size_t node = (size_t)(row0 + r + half * 8);
      float o = acc2[r] * s2 + c2;
      o = o > 0.0f ? o : 0.0f;
      if (resid) o += resid[node * HID + n];
      hout[node * HID + n] = o;
    }
  }
}

// ---------------------------------------------------------------------------
// Global mean pool: sums into d_out (pre-zeroed), counts into cnts
// ---------------------------------------------------------------------------
__global__ void pool_sum_kernel(const float* __restrict__ h,
                                const int* __restrict__ batch,
                                float* __restrict__ sums,   // [G, 128]
                                float* __restrict__ cnts,   // [G]
                                long long total)            // N * 128
{
  long long i = (long long)blockIdx.x * blockDim.x + threadIdx.x;
  if (i < total) {
    const int node = (int)(i >> 7);
    const int c    = (int)(i & 127);
    const int g    = batch[node];
    __hip_atomic_fetch_add(&sums[(size_t)g * HID + c], h[i],
                           __ATOMIC_RELAXED, __HIP_MEMORY_SCOPE_AGENT);
    if (c == 0)
      __hip_atomic_fetch_add(&cnts[g], 1.0f,
                             __ATOMIC_RELAXED, __HIP_MEMORY_SCOPE_AGENT);
  }
}

__global__ void pool_div_kernel(float* __restrict__ out,
                                const float* __restrict__ cnts,
                                long long total)            // G * 128
{
  long long i = (long long)blockIdx.x * blockDim.x + threadIdx.x;
  if (i < total) out[i] /= fmaxf(cnts[i >> 7], 1.0f);
}

// ---------------------------------------------------------------------------
// launch
// ---------------------------------------------------------------------------
extern "C" void kernel_launch(void* const* d_in, const int* in_sizes, int n_in,
                              void* d_out, int out_size, void* d_ws, size_t ws_size,
                              hipStream_t stream) {
  const float* x        = (const float*)d_in[0];   // [N, 64]
  const float* ea       = (const float*)d_in[1];   // [E, 16]
  const int*   eidx     = (const int*)  d_in[2];   // [2, E]
  const int*   batch    = (const int*)  d_in[3];   // [N]
  const float* m0w1     = (const float*)d_in[4];
  const float* m0b1     = (const float*)d_in[5];
  const float* m0bg     = (const float*)d_in[6];
  const float* m0bb     = (const float*)d_in[7];
  const float* m0bm     = (const float*)d_in[8];
  const float* m0bv     = (const float*)d_in[9];
  const float* m0w2     = (const float*)d_in[10];
  const float* m0b2     = (const float*)d_in[11];
  const float* e0w      = (const float*)d_in[12];  // [16, 64]
  const float* e0b      = (const float*)d_in[13];  // [64]
  const float* msw1     = (const float*)d_in[14];
  const float* msb1     = (const float*)d_in[15];
  const float* msbg     = (const float*)d_in[16];
  const float* msbb     = (const float*)d_in[17];
  const float* msbm     = (const float*)d_in[18];
  const float* msbv     = (const float*)d_in[19];
  const float* msw2     = (const float*)d_in[20];
  const float* msb2     = (const float*)d_in[21];
  const float* ew       = (const float*)d_in[22];  // [2, 16, 128]
  const float* ebias    = (const float*)d_in[23];  // [2, 128]
  const float* bng      = (const float*)d_in[24];  // [3, 128]
  const float* bnb      = (const float*)d_in[25];
  const float* bnm      = (const float*)d_in[26];
  const float* bnv      = (const float*)d_in[27];

  const int N = in_sizes[0] / 64;          // 100000
  const int E = in_sizes[1] / EDGE_DIM;    // 1600000
  const int G = out_size / HID;            // 2048

  const int* srcI = eidx;
  const int* dstI = eidx + E;

  // workspace layout
  const size_t hbytes = (size_t)N * HID * sizeof(float);
  float* hA   = (float*)d_ws;
  float* hB   = (float*)((char*)d_ws + hbytes);
  float* agg  = (float*)((char*)d_ws + 2 * hbytes);
  float* cnts = (float*)((char*)d_ws + 3 * hbytes);
  float* outF = (float*)d_out;

  const int EDGE_BLOCKS = 4096;            // 8 waves/block, grid-stride
  const int MLP_BLOCKS  = N / 16;          // N divisible by 16
  auto zgrid = [](long long n) { return (unsigned)((n + 255) / 256); };

  // ---------------- layer 0 (Kin = 64, no residual) ----------------
  zero_f32_kernel<<<zgrid((long long)N * 64), 256, 0, stream>>>(agg, (long long)N * 64);
  gine_edge_kernel<<<EDGE_BLOCKS, 256, 0, stream>>>(
      x, ea, srcI, dstI, e0w, e0b, agg, 64, E);
  gine_mlp_wmma_kernel<<<MLP_BLOCKS, 256, 0, stream>>>(
      x, agg, 64, m0w1, m0b1, m0bg, m0bb, m0bm, m0bv, m0w2, m0b2,
      bng + 0 * HID, bnb + 0 * HID, bnm + 0 * HID, bnv + 0 * HID,
      nullptr, hA);

  // ---------------- layer 1 (Kin = 128, residual) ----------------
  zero_f32_kernel<<<zgrid((long long)N * HID), 256, 0, stream>>>(agg, (long long)N * HID);
  gine_edge_kernel<<<EDGE_BLOCKS, 256, 0, stream>>>(
      hA, ea, srcI, dstI, ew + 0 * EDGE_DIM * HID, ebias + 0 * HID, agg, HID, E);
  gine_mlp_wmma_kernel<<<MLP_BLOCKS, 256, 0, stream>>>(
      hA, agg, HID, msw1, msb1, msbg, msbb, msbm, msbv, msw2, msb2,
      bng + 1 * HID, bnb + 1 * HID, bnm + 1 * HID, bnv + 1 * HID,
      hA, hB);

  // ---------------- layer 2 (Kin = 128, residual) ----------------
  zero_f32_kernel<<<zgrid((long long)N * HID), 256, 0, stream>>>(agg, (long long)N * HID);
  gine_edge_kernel<<<EDGE_BLOCKS, 256, 0, stream>>>(
      hB, ea, srcI, dstI, ew + 1 * EDGE_DIM * HID, ebias + 1 * HID, agg, HID, E);
  gine_mlp_wmma_kernel<<<MLP_BLOCKS, 256, 0, stream>>>(
      hB, agg, HID, msw1, msb1, msbg, msbb, msbm, msbv, msw2, msb2,
      bng + 2 * HID, bnb + 2 * HID, bnm + 2 * HID, bnv + 2 * HID,
      hB, hA);

  // ---------------- global mean pool ----------------
  zero_f32_kernel<<<zgrid((long long)G * HID), 256, 0, stream>>>(outF, (long long)G * HID);
  zero_f32_kernel<<<zgrid(G), 256, 0, stream>>>(cnts, (long long)G);
  pool_sum_kernel<<<zgrid((long long)N * HID), 256, 0, stream>>>(
      hA, batch, outF, cnts, (long long)N * HID);
  pool_div_kernel<<<zgrid((long long)G * HID), 256, 0, stream>>>(
      outF, cnts, (long long)G * HID);
}